// Q_MultiHeadedAttention_28793460753159
// MI455X (gfx1250) — compile-verified
//
#include <hip/hip_runtime.h>

// ---------------------------------------------------------------------------
// Quantum multi-headed attention (SEQ=4, NQ=3, DIM=8) collapsed to its
// algebraic core: rank-1 density matrices -> diag/trace-only observables.
// Single wave32 workgroup; the 16 U rho U^dag conjugations run through
// v_wmma_f32_16x16x4_f32 via the real 16x16 embedding of complex 8x8.
// ---------------------------------------------------------------------------

typedef float v2f __attribute__((ext_vector_type(2)));
typedef float v8f __attribute__((ext_vector_type(8)));

#define W_MUL 0.6324555320336759f  // sqrt(2)*5^-0.5 (he_std * lrmul)
#define TWO_PI 6.283185307179586f

struct Cplx { float re, im; };
__device__ __forceinline__ Cplx cmul(Cplx a, Cplx b) {
  return { a.re * b.re - a.im * b.im, a.re * b.im + a.im * b.re };
}

// 2x2 complex  rz(pz) @ ry(py) @ rx(px)
__device__ __forceinline__ void gate2(float px, float py, float pz, Cplx g[2][2]) {
  float cx = cosf(px * 0.5f), sx = sinf(px * 0.5f);
  float cy = cosf(py * 0.5f), sy = sinf(py * 0.5f);
  float cz = cosf(pz * 0.5f), sz = sinf(pz * 0.5f);
  // ry@rx
  Cplx m00 = {  cy * cx,  sy * sx };
  Cplx m01 = { -sy * cx, -cy * sx };
  Cplx m10 = {  sy * cx, -cy * sx };
  Cplx m11 = {  cy * cx, -sy * sx };
  Cplx e0 = { cz, -sz };   // e^{-i pz/2}
  Cplx e1 = { cz,  sz };   // e^{+i pz/2}
  g[0][0] = cmul(e0, m00); g[0][1] = cmul(e0, m01);
  g[1][0] = cmul(e1, m10); g[1][1] = cmul(e1, m11);
}

__global__ __launch_bounds__(32, 1)
void qattn_kernel(const float* __restrict__ q, const float* __restrict__ k,
                  const float* __restrict__ v, const float* __restrict__ wq,
                  const float* __restrict__ wk, const float* __restrict__ wv,
                  float* __restrict__ out) {
  const int lane = threadIdx.x;  // one wave32

  __shared__ Cplx  sg[9][2][2];          // 9 per-qubit 2x2 gates (Uq,Uk,Uv)
  __shared__ float sU[3][8][8][2];       // 8x8 complex unitaries
  __shared__ float sxdsn[3][4][8];       // normalized |row| encodings
  __shared__ float svec[3][4][8][2];     // U @ xdsn (rank-1 state vectors)
  __shared__ float sdq[4][8], sdk[4][8]; // |qvec|^2, |kvec|^2 (diagonals)
  __shared__ Cplx  sup[16][2][2];        // per-(i,j) value-rotation 2x2
  __shared__ float sUphi[16][16][16];    // R(U_phi): real 16x16 embedding
  __shared__ float sRho[4][16][16];      // R(outer(vvec, conj(vvec)))
  __shared__ float sT[16][16];           // matmul temp
  __shared__ float sA[16][16][16];       // R(U rho U^T) results

  // ---- Phase 1: 9 per-qubit gates -------------------------------------
  if (lane < 9) {
    int u = lane / 3, t = lane % 3;
    const float* w = (u == 0) ? wq : (u == 1) ? wk : wv;
    Cplx g[2][2];
    gate2(w[3 * t + 0] * W_MUL, w[3 * t + 1] * W_MUL, w[3 * t + 2] * W_MUL, g);
    sg[lane][0][0] = g[0][0]; sg[lane][0][1] = g[0][1];
    sg[lane][1][0] = g[1][0]; sg[lane][1][1] = g[1][1];
  }
  __syncthreads();

  // ---- Phase 2: kron -> 8x8 unitaries; Phase 3: encodings -------------
  for (int e = lane; e < 192; e += 32) {
    int u = e >> 6, rc = e & 63, r = rc >> 3, c = rc & 7;
    Cplx p = sg[u * 3 + 0][(r >> 2) & 1][(c >> 2) & 1];
    p = cmul(p, sg[u * 3 + 1][(r >> 1) & 1][(c >> 1) & 1]);
    p = cmul(p, sg[u * 3 + 2][r & 1][c & 1]);
    sU[u][r][c][0] = p.re; sU[u][r][c][1] = p.im;
  }
  if (lane < 12) {
    int m = lane >> 2, i = lane & 3;
    const float* src = (m == 0) ? q : (m == 1) ? k : v;
    float tmp[8], ss = 0.f;
    for (int x = 0; x < 8; ++x) { float a = fabsf(src[i * 8 + x]); tmp[x] = a; ss += a * a; }
    float inv = 1.0f / (sqrtf(ss) + 1e-12f);
    for (int x = 0; x < 8; ++x) sxdsn[m][i][x] = tmp[x] * inv;
  }
  __syncthreads();

  // ---- Phase 4: state vectors vec = U @ xdsn --------------------------
  for (int e = lane; e < 96; e += 32) {
    int m = e / 32, rem = e % 32, i = rem >> 3, r = rem & 7;
    float re = 0.f, im = 0.f;
    for (int c = 0; c < 8; ++c) {
      float x = sxdsn[m][i][c];
      re += sU[m][r][c][0] * x; im += sU[m][r][c][1] * x;
    }
    svec[m][i][r][0] = re; svec[m][i][r][1] = im;
  }
  __syncthreads();

  // ---- Phase 5: density diagonals -------------------------------------
  for (int e = lane; e < 64; e += 32) {
    int m = e >> 5, rem = e & 31, i = rem >> 3, x = rem & 7;
    float re = svec[m][i][x][0], im = svec[m][i][x][1];
    float val = re * re + im * im;
    if (m == 0) sdq[i][x] = val; else sdk[i][x] = val;
  }
  __syncthreads();

  // ---- Phase 6: factored Z-measurements -> mean score -> phi gate -----
  if (lane < 16) {
    int i = lane >> 2, j = lane & 3;
    float Zq[3] = {0.f, 0.f, 0.f}, Zk[3] = {0.f, 0.f, 0.f}, Sk = 0.f;
    for (int x = 0; x < 8; ++x) {
      float a = sdq[i][x], b = sdk[j][x];
      Sk += b;
      for (int t = 0; t < 3; ++t) {
        float s = ((x >> (2 - t)) & 1) ? -1.f : 1.f;
        Zq[t] += s * a; Zk[t] += s * b;
      }
    }
    float acc = 0.f;
    for (int t = 0; t < 3; ++t) {
      acc += (Zq[t] * Sk + 1.f) * 0.5f;      // q-register bits (CNOT controls)
      acc += (Zq[t] * Zk[t] + 1.f) * 0.5f;   // k-register bits (XORed targets)
    }
    float phi = (acc / 6.f - 0.5f) * TWO_PI;
    Cplx g[2][2];
    gate2(phi * 0.5f, phi * 0.5f, phi, g);   // rz(phi) ry(phi/2) rx(phi/2)
    sup[lane][0][0] = g[0][0]; sup[lane][0][1] = g[0][1];
    sup[lane][1][0] = g[1][0]; sup[lane][1][1] = g[1][1];
  }
  __syncthreads();

  // ---- Phase 7: real 16x16 embeddings  R(M)=[[Re,-Im],[Im,Re]] --------
  for (int e = lane; e < 16 * 256; e += 32) {
    int p = e >> 8, rc = e & 255, r16 = rc >> 4, c16 = rc & 15;
    int r = r16 & 7, c = c16 & 7;
    Cplx z = sup[p][(r >> 2) & 1][(c >> 2) & 1];
    z = cmul(z, sup[p][(r >> 1) & 1][(c >> 1) & 1]);
    z = cmul(z, sup[p][r & 1][c & 1]);
    sUphi[p][r16][c16] = (r16 < 8) ? ((c16 < 8) ? z.re : -z.im)
                                   : ((c16 < 8) ? z.im :  z.re);
  }
  for (int e = lane; e < 4 * 256; e += 32) {
    int j = e >> 8, rc = e & 255, r16 = rc >> 4, c16 = rc & 15;
    int r = r16 & 7, c = c16 & 7;
    float ar = svec[2][j][r][0], ai = svec[2][j][r][1];
    float br = svec[2][j][c][0], bi = svec[2][j][c][1];
    float zre = ar * br + ai * bi;           // (a conj(b)).re
    float zim = ai * br - ar * bi;           // (a conj(b)).im
    sRho[j][r16][c16] = (r16 < 8) ? ((c16 < 8) ? zre : -zim)
                                  : ((c16 < 8) ? zim :  zre);
  }
  __syncthreads();

  // ---- Phase 8: A_p = Uphi_p @ Rho_{p&3} @ Uphi_p^T via f32 WMMA ------
  const int mA = lane & 15;        // A-frag row (ISA 16x4 layout)
  const int kh = lane >> 4;        // A/B K-half select
  const int nC = lane & 15;        // C-frag column
  const int mC = (lane >> 4) * 8;  // C-frag row base
  for (int p = 0; p < 16; ++p) {
    const int j = p & 3;
#if defined(__gfx1250__)
    v8f c = { 0.f, 0.f, 0.f, 0.f, 0.f, 0.f, 0.f, 0.f };
    for (int kk = 0; kk < 16; kk += 4) {
      v2f a, b;
      a.x = sUphi[p][mA][kk + kh * 2 + 0];
      a.y = sUphi[p][mA][kk + kh * 2 + 1];
      b.x = sRho[j][kk + kh * 2 + 0][nC];
      b.y = sRho[j][kk + kh * 2 + 1][nC];
      c = __builtin_amdgcn_wmma_f32_16x16x4_f32(false, a, false, b,
                                                (short)0, c, false, false);
    }
    for (int r = 0; r < 8; ++r) sT[mC + r][nC] = c[r];
    __syncthreads();
    v8f c2 = { 0.f, 0.f, 0.f, 0.f, 0.f, 0.f, 0.f, 0.f };
    for (int kk = 0; kk < 16; kk += 4) {
      v2f a, b;
      a.x = sT[mA][kk + kh * 2 + 0];
      a.y = sT[mA][kk + kh * 2 + 1];
      b.x = sUphi[p][nC][kk + kh * 2 + 0];   // B = Uphi^T
      b.y = sUphi[p][nC][kk + kh * 2 + 1];
      c2 = __builtin_amdgcn_wmma_f32_16x16x4_f32(false, a, false, b,
                                                 (short)0, c2, false, false);
    }
    for (int r = 0; r < 8; ++r) sA[p][mC + r][nC] = c2[r];
    __syncthreads();
#else
    // Host pass / non-gfx1250 fallback: plain FMA matmuls.
    for (int e = lane; e < 256; e += 32) {
      int r = e >> 4, cc = e & 15; float acc = 0.f;
      for (int x = 0; x < 16; ++x) acc += sUphi[p][r][x] * sRho[j][x][cc];
      sT[r][cc] = acc;
    }
    __syncthreads();
    for (int e = lane; e < 256; e += 32) {
      int r = e >> 4, cc = e & 15; float acc = 0.f;
      for (int x = 0; x < 16; ++x) acc += sT[r][x] * sUphi[p][cc][x];
      sA[p][r][cc] = acc;
    }
    __syncthreads();
#endif
  }

  // ---- Phase 9: out[i][t] = (prod traces * Z_t(diag A_{i,0}) + 1)/2 ---
  if (lane < 12) {
    int i = lane / 3, t = lane % 3;
    float T = 1.0f;
    for (int j = 1; j < 4; ++j) {
      float tr = 0.f;
      for (int m = 0; m < 8; ++m) tr += sA[i * 4 + j][m][m];
      T *= tr;
    }
    float acc = 0.f;
    for (int m = 0; m < 8; ++m) {
      float s = ((m >> (2 - t)) & 1) ? -1.f : 1.f;
      acc += s * sA[i * 4 + 0][m][m];
    }
    out[i * 3 + t] = (T * acc + 1.f) * 0.5f;
  }
}

extern "C" void kernel_launch(void* const* d_in, const int* in_sizes, int n_in,
                              void* d_out, int out_size, void* d_ws, size_t ws_size,
                              hipStream_t stream) {
  (void)in_sizes; (void)n_in; (void)out_size; (void)d_ws; (void)ws_size;
  qattn_kernel<<<1, 32, 0, stream>>>(
      (const float*)d_in[0], (const float*)d_in[1], (const float*)d_in[2],
      (const float*)d_in[3], (const float*)d_in[4], (const float*)d_in[5],
      (float*)d_out);
}